// YOLOV1Loss_65008624992995
// MI455X (gfx1250) — compile-verified
//
#include <hip/hip_runtime.h>

typedef __attribute__((ext_vector_type(2))) float v2f;
typedef __attribute__((ext_vector_type(8))) float v8f;

namespace {
constexpr int   kS       = 7;
constexpr int   kN       = 30;                  // 5*B + C
constexpr int   kBatch   = 16384;
constexpr int   kCells   = kBatch * kS * kS;    // 802816
constexpr float kCoord   = 5.0f;
constexpr float kNoobj   = 0.5f;
constexpr int   kThreads = 256;                 // 8 waves (wave32)
constexpr int   kBlocks  = kCells / kThreads;   // 3136, exact
}

// Wave32 reduction through the matrix pipe: D = A(16x4, lane partials) x
// B(4x16, all ones) + 0.  Every D column N holds rowsum contributions; lane L
// (<16) holds rows 0..7 of column L and lane L+16 holds rows 8..15, so one
// xor-16 shuffle after the per-lane 8-element sum yields the full wave total
// in f32 precision.  Then LDS across the 8 waves of the block.
__device__ __forceinline__ float block_reduce_wmma(float acc, float* lds) {
  v2f a; a.x = acc;  a.y = 0.0f;
  v2f b; b.x = 1.0f; b.y = 1.0f;
  v8f c = {};
  v8f d = __builtin_amdgcn_wmma_f32_16x16x4_f32(false, a, false, b,
                                                (short)0, c, false, false);
  float s = ((d[0] + d[1]) + (d[2] + d[3])) + ((d[4] + d[5]) + (d[6] + d[7]));
  s += __shfl_xor(s, 16, 32);
  const int lane = threadIdx.x & 31;
  const int wid  = threadIdx.x >> 5;
  if (lane == 0) lds[wid] = s;
  __syncthreads();
  float tot = 0.0f;
  if (threadIdx.x == 0) {
#pragma unroll
    for (int i = 0; i < kThreads / 32; ++i) tot += lds[i];
  }
  return tot;  // valid in thread 0
}

__global__ void __launch_bounds__(kThreads)
yolo_v1_loss_cells(const float* __restrict__ x, const float* __restrict__ y,
                   float* __restrict__ partials) {
  __shared__ float lds[kThreads / 32];
  const int cell = blockIdx.x * kThreads + threadIdx.x;  // always < kCells

  const float2* px2 = reinterpret_cast<const float2*>(x) + (size_t)cell * (kN / 2);
  const float2* py2 = reinterpret_cast<const float2*>(y) + (size_t)cell * (kN / 2);

  // Box channels [0..9]: keep in registers.
  float xb[10], yb[10];
#pragma unroll
  for (int i = 0; i < 5; ++i) {
    float2 t = px2[i]; xb[2 * i] = t.x; xb[2 * i + 1] = t.y;
    float2 u = py2[i]; yb[2 * i] = u.x; yb[2 * i + 1] = u.y;
  }
  // Class channels [10..29]: stream and accumulate, never kept live.
  float loss_cls = 0.0f;
#pragma unroll
  for (int i = 5; i < 15; ++i) {
    float2 t = px2[i];
    float2 u = py2[i];
    float d0 = t.x - u.x, d1 = t.y - u.y;
    loss_cls += d0 * d0 + d1 * d1;
  }

  const float invS   = 1.0f / (float)kS;
  const float objf   = (yb[4] > 0.0f) ? 1.0f : 0.0f;
  const float noobjf = (yb[4] == 0.0f) ? 1.0f : 0.0f;

  // Target xyxy from target box 0 only (matches reference).
  const float tx0 = yb[0] * invS - 0.5f * yb[2];
  const float ty0 = yb[1] * invS - 0.5f * yb[3];
  const float tx1 = yb[0] * invS + 0.5f * yb[2];
  const float ty1 = yb[1] * invS + 0.5f * yb[3];
  const float area_t = (tx1 - tx0) * (ty1 - ty0);

  float iou0, iou1;
  {
    float px0 = xb[0] * invS - 0.5f * xb[2], py0 = xb[1] * invS - 0.5f * xb[3];
    float px1 = xb[0] * invS + 0.5f * xb[2], py1 = xb[1] * invS + 0.5f * xb[3];
    float iw = fmaxf(fminf(px1, tx1) - fmaxf(px0, tx0), 0.0f);
    float ih = fmaxf(fminf(py1, ty1) - fmaxf(py0, ty0), 0.0f);
    float inter  = iw * ih;
    float area_p = (px1 - px0) * (py1 - py0);
    iou0 = inter / (area_p + area_t - inter);
  }
  {
    float px0 = xb[5] * invS - 0.5f * xb[7], py0 = xb[6] * invS - 0.5f * xb[8];
    float px1 = xb[5] * invS + 0.5f * xb[7], py1 = xb[6] * invS + 0.5f * xb[8];
    float iw = fmaxf(fminf(px1, tx1) - fmaxf(px0, tx0), 0.0f);
    float ih = fmaxf(fminf(py1, ty1) - fmaxf(py0, ty0), 0.0f);
    float inter  = iw * ih;
    float area_p = (px1 - px0) * (py1 - py0);
    iou1 = inter / (area_p + area_t - inter);
  }

  // argmax over B=2; ties pick box 0 (jnp.argmax semantics).
  const bool  r1      = iou1 > iou0;
  const float max_iou = r1 ? iou1 : iou0;

  const float prx = r1 ? xb[5] : xb[0];
  const float pry = r1 ? xb[6] : xb[1];
  const float prw = r1 ? xb[7] : xb[2];
  const float prh = r1 ? xb[8] : xb[3];
  const float prc = r1 ? xb[9] : xb[4];
  const float trx = r1 ? yb[5] : yb[0];
  const float trY = r1 ? yb[6] : yb[1];
  const float trw = r1 ? yb[7] : yb[2];
  const float trh = r1 ? yb[8] : yb[3];

  const float dx = prx - trx, dy = pry - trY;
  const float loss_xy = dx * dx + dy * dy;
  const float dw = sqrtf(prw) - sqrtf(trw);
  const float dh = sqrtf(prh) - sqrtf(trh);
  const float loss_wh = dw * dw + dh * dh;
  const float dob = prc - max_iou;
  const float loss_obj = dob * dob;
  const float d4 = xb[4] - yb[4], d9 = xb[9] - yb[9];
  const float loss_noobj = d4 * d4 + d9 * d9;

  float acc = objf * (kCoord * (loss_xy + loss_wh) + loss_obj + loss_cls) +
              noobjf * (kNoobj * loss_noobj);
  acc *= (1.0f / (float)kBatch);

  const float tot = block_reduce_wmma(acc, lds);
  if (threadIdx.x == 0) partials[blockIdx.x] = tot;
}

__global__ void __launch_bounds__(kThreads)
yolo_v1_loss_reduce(const float* __restrict__ partials, float* __restrict__ out) {
  __shared__ float lds[kThreads / 32];
  float acc = 0.0f;
  for (int i = threadIdx.x; i < kBlocks; i += kThreads) acc += partials[i];
  const float tot = block_reduce_wmma(acc, lds);
  if (threadIdx.x == 0) out[0] = tot;
}

extern "C" void kernel_launch(void* const* d_in, const int* in_sizes, int n_in,
                              void* d_out, int out_size, void* d_ws, size_t ws_size,
                              hipStream_t stream) {
  (void)in_sizes; (void)n_in; (void)out_size; (void)ws_size;
  const float* x = (const float*)d_in[0];
  const float* y = (const float*)d_in[1];
  float* out      = (float*)d_out;
  float* partials = (float*)d_ws;  // kBlocks floats = 12.25 KB

  yolo_v1_loss_cells<<<kBlocks, kThreads, 0, stream>>>(x, y, partials);
  yolo_v1_loss_reduce<<<1, kThreads, 0, stream>>>(partials, out);
}